// VolumeRenderer_4243427689048
// MI455X (gfx1250) — compile-verified
//
#include <hip/hip_runtime.h>
#include <math.h>

// Volume ray-marching (Plenoxels-style) for MI455X / gfx1250.
// One wave32 lane per ray; 192 DDA steps; nearest-voxel float4 gathers from a
// 32MB grid that is L2-resident (192MB L2). Latency-bound -> deep software
// pipeline: the ray trajectory is voxel-independent, so 8 gathers per chunk
// are issued before any is consumed (MLP=8 per wave), then the exp/sigmoid
// compositing consumes them while the next chunk's first voxel is prefetched.

#define RGRID   128
#define NSTEPS  192
#define STEPSZ  0.001f
#define BACKGND 1.0f
#define CHUNK   8      // 192 / 8 = 24 chunks; 8 in-flight global_load_b128

__device__ __forceinline__ float fast_sigmoid(float x) {
    // 1/(1+exp(-x)) with hardware v_exp_f32 + v_rcp_f32 (TRANS ops on CDNA5)
    float e = __expf(-x);
    return __builtin_amdgcn_rcpf(1.0f + e);
}

__global__ __launch_bounds__(256)
void volrend_kernel(const float* __restrict__ grid,
                    const float* __restrict__ origins,
                    const float* __restrict__ dirs,
                    float* __restrict__ out,
                    int B)
{
    const int ray = blockIdx.x * blockDim.x + threadIdx.x;
    if (ray >= B) return;

    // ---- per-ray setup (all-TRANS: v_rsq_f32 / v_rcp_f32, no precise-div
    // fixup sequences) --------------------------------------------------------
    const float ox = origins[3 * ray + 0];
    const float oy = origins[3 * ray + 1];
    const float oz = origins[3 * ray + 2];

    float dx = dirs[3 * ray + 0];
    float dy = dirs[3 * ray + 1];
    float dz = dirs[3 * ray + 2];
    {
        const float inv_n = __builtin_amdgcn_rsqf(dx * dx + dy * dy + dz * dz);
        dx *= inv_n; dy *= inv_n; dz *= inv_n;
    }
    const float ivx = __builtin_amdgcn_rcpf(dx + 1e-9f);
    const float ivy = __builtin_amdgcn_rcpf(dy + 1e-9f);
    const float ivz = __builtin_amdgcn_rcpf(dz + 1e-9f);

    // entry/exit of the unit cube (_dda_unit on origins)
    float t, tmax;
    {
        const float a1x = -ox * ivx, a2x = a1x + ivx;
        const float a1y = -oy * ivy, a2y = a1y + ivy;
        const float a1z = -oz * ivz, a2z = a1z + ivz;
        const float lox = fminf(a1x, a2x), hix = fmaxf(a1x, a2x);
        const float loy = fminf(a1y, a2y), hiy = fmaxf(a1y, a2y);
        const float loz = fminf(a1z, a2z), hiz = fmaxf(a1z, a2z);
        t    = fmaxf(fmaxf(lox, fmaxf(loy, loz)), 0.0f);
        tmax = fminf(fminf(hix, fminf(hiy, hiz)), 1.0e9f);
    }

    const float4* __restrict__ g4 = (const float4*)grid;
    const float invR = 1.0f / (float)RGRID;

    float light = 1.0f;
    float accr = 0.0f, accg = 0.0f, accb = 0.0f;

    bool alive = (t < tmax);

    // ---- main march: chunks of CHUNK steps, loads pipelined ahead of shading
    for (int s0 = 0; s0 < NSTEPS && alive; s0 += CHUNK) {
        float  dts[CHUNK];
        float4 vox[CHUNK];
        float  tt = t;

        // Phase 1: pure-geometry DDA walk (voxel-independent!) — issue all
        // CHUNK gathers before any of them is consumed.
#pragma unroll
        for (int j = 0; j < CHUNK; ++j) {
            const bool v = (tt < tmax);

            const float px = fmaf(tt, dx, ox) * (float)RGRID;
            const float py = fmaf(tt, dy, oy) * (float)RGRID;
            const float pz = fmaf(tt, dz, oz) * (float)RGRID;

            // clamp in float domain: one cvt per axis instead of two
            const float fx = fminf(fmaxf(floorf(px), 0.0f), (float)(RGRID - 1));
            const float fy = fminf(fmaxf(floorf(py), 0.0f), (float)(RGRID - 1));
            const float fz = fminf(fmaxf(floorf(pz), 0.0f), (float)(RGRID - 1));
            const int ix = (int)fx;
            const int iy = (int)fy;
            const int iz = (int)fz;

            const float cx = px - fx;
            const float cy = py - fy;
            const float cz = pz - fz;

            // _dda_unit(pos_t, invdirs)
            const float b1x = -cx * ivx, b2x = b1x + ivx;
            const float b1y = -cy * ivy, b2y = b1y + ivy;
            const float b1z = -cz * ivz, b2z = b1z + ivz;
            const float lox = fminf(b1x, b2x), hix = fmaxf(b1x, b2x);
            const float loy = fminf(b1y, b2y), hiy = fmaxf(b1y, b2y);
            const float loz = fminf(b1z, b2z), hiz = fmaxf(b1z, b2z);
            const float smin = fmaxf(fmaxf(lox, fmaxf(loy, loz)), 0.0f);
            const float smax = fminf(fminf(hix, fminf(hiy, hiz)), 1.0e9f);

            const float dt = (smax - smin) * invR + STEPSZ;
            dts[j] = dt;

            const int lin = (ix * RGRID + iy) * RGRID + iz;

            // Predicated gather (global_load_b128). Dead lanes take a zero
            // voxel: sigma=0 -> att=1 -> weight=0 -> exact no-op update.
            if (v) {
                vox[j] = g4[lin];
            } else {
                vox[j] = make_float4(0.0f, 0.0f, 0.0f, 0.0f);
            }

            // Freeze t for dead lanes (keeps everything finite and matches
            // the reference's masking; dt >= STEPSZ so dead lanes stay dead).
            tt = v ? (tt + dt) : tt;
        }

        // Warm L2/L0 for the next chunk's first voxel while we shade
        // (global_prefetch_b8 on gfx1250).
        if (tt < tmax) {
            const float px = fmaf(tt, dx, ox) * (float)RGRID;
            const float py = fmaf(tt, dy, oy) * (float)RGRID;
            const float pz = fmaf(tt, dz, oz) * (float)RGRID;
            const float fx = fminf(fmaxf(floorf(px), 0.0f), (float)(RGRID - 1));
            const float fy = fminf(fmaxf(floorf(py), 0.0f), (float)(RGRID - 1));
            const float fz = fminf(fmaxf(floorf(pz), 0.0f), (float)(RGRID - 1));
            const int lin = ((int)fx * RGRID + (int)fy) * RGRID + (int)fz;
            __builtin_prefetch((const void*)(g4 + lin), 0, 0);
        }

        // Phase 2: compositing (exp/sigmoid heavy) — consumes the in-flight
        // loads; serial only through `light`.
#pragma unroll
        for (int j = 0; j < CHUNK; ++j) {
            const float sigma = fmaxf(vox[j].w, 0.0f);
            const float att   = __expf(-dts[j] * sigma);
            const float w     = light * (1.0f - att);
            accr = fmaf(w, fast_sigmoid(vox[j].x), accr);
            accg = fmaf(w, fast_sigmoid(vox[j].y), accg);
            accb = fmaf(w, fast_sigmoid(vox[j].z), accb);
            light *= att;
        }

        t = tt;
        alive = (t < tmax);
    }

    out[3 * ray + 0] = accr + BACKGND * light;
    out[3 * ray + 1] = accg + BACKGND * light;
    out[3 * ray + 2] = accb + BACKGND * light;
}

extern "C" void kernel_launch(void* const* d_in, const int* in_sizes, int n_in,
                              void* d_out, int out_size, void* d_ws, size_t ws_size,
                              hipStream_t stream) {
    const float* grid    = (const float*)d_in[0];  // (128,128,128,4) f32
    const float* origins = (const float*)d_in[1];  // (B,3) f32
    const float* dirs    = (const float*)d_in[2];  // (B,3) f32
    // d_in[3] = viewdirs: unused by the reference
    float* out = (float*)d_out;                    // (B,3) f32

    const int B = in_sizes[1] / 3;
    const int block = 256;                         // 8 wave32 waves
    const int ngrid = (B + block - 1) / block;

    volrend_kernel<<<ngrid, block, 0, stream>>>(grid, origins, dirs, out, B);
}